// SAGE_58471684767918
// MI455X (gfx1250) — compile-verified
//
#include <hip/hip_runtime.h>
#include <math.h>

// Problem constants from the reference setup_inputs()
#define N0c 500000
#define N1c 50000     // 3125 * 16  -> no row guards needed in GEMM0
#define N2c 4096      //  256 * 16  -> no row guards needed in GEMM1
#define E0c 1280000
#define E1c 40960
#define INc 128
#define HIDc 256
#define OUTc 47
#define OUTP 64       // layer-1 N padded to one 64-wide block tile

typedef float v2f __attribute__((ext_vector_type(2)));
typedef float v8f __attribute__((ext_vector_type(8)));

// ---------------------------------------------------------------- zero scratch
__global__ void zero_f4_kernel(float4* __restrict__ p, int n4) {
    int t = blockIdx.x * blockDim.x + threadIdx.x;
    if (t < n4) p[t] = make_float4(0.f, 0.f, 0.f, 0.f);
}

// ------------------------------------------- pad [rows x 47] -> [rows x 64]
__global__ void pad_cols64_kernel(const float* __restrict__ src,
                                  float* __restrict__ dst,
                                  int rows, int cs) {
    int t = blockIdx.x * blockDim.x + threadIdx.x;
    if (t >= rows * 64) return;
    int r = t >> 6, c = t & 63;
    dst[t] = (c < cs) ? src[r * cs + c] : 0.0f;
}

// ------------------------------------------------- scatter-add (mean, part 1)
// One edge handled by (ld/4) lanes, each lane a float4 slice of the source row.
// The accumulators (25.6 MB / 4 MB) are L2-resident on MI455X (192 MB L2), so
// the f32 atomic RMW traffic never leaves the chip; only the 512B-granule
// random row gathers (~655 MB total, ~30 us at 23.3 TB/s) hit HBM. This phase
// is the roofline-dominant cost of the whole network.
__global__ void scatter_mean_accum(const float* __restrict__ xsrc,
                                   const int* __restrict__ src,
                                   const int* __restrict__ dst,
                                   float* __restrict__ agg,
                                   float* __restrict__ cnt,
                                   int E, int ld, int cshift) {
    int t = blockIdx.x * blockDim.x + threadIdx.x;
    int e = t >> cshift;
    if (e >= E) return;
    int c = (t & ((1 << cshift) - 1)) << 2;
    int s = src[e];
    int d = dst[e];
    const float4 v = *(const float4*)(xsrc + (size_t)s * ld + c);
    float* p = agg + (size_t)d * ld + c;
    atomicAdd(p + 0, v.x);
    atomicAdd(p + 1, v.y);
    atomicAdd(p + 2, v.z);
    atomicAdd(p + 3, v.w);
    if (c == 0) atomicAdd(cnt + d, 1.0f);
}

// ------------------------------------------------------ divide by bucket size
__global__ void div_cnt_kernel(float* __restrict__ agg,
                               const float* __restrict__ cnt,
                               int total, int dshift) {
    int t = blockIdx.x * blockDim.x + threadIdx.x;
    if (t >= total) return;
    agg[t] = agg[t] / fmaxf(cnt[t >> dshift], 1.0f);
}

// ------------------------------------------------------------- fused SAGE GEMM
// C[M,NLD] = act( A1[M,K1] @ B1[K1,NLD] + A2[M,K2] @ B2[K2,NLD] + bias[NLD] )
// Compile-time K1,K2,NLD; M % 16 == 0 and all operands padded to NLD columns,
// so the kernel is fully guard-free: EXEC is all-ones everywhere (WMMA
// requirement) and the unrolled MAC loop is pure
//     ds_load_2addr_b32 -> 2x global_load_b32 -> v_wmma_f32_16x16x4_f32
// with immediate address offsets.
//
// Block = 128 threads (4 wave32); block tile = 16 rows x 64 cols; wave w owns
// cols [blockIdx.y*64 + w*16, +16). Fragment layouts per CDNA5 ISA:
//   A 16x4 f32 : lane M = lane&15 ; VGPR0 = K=(lane>=16?2:0), VGPR1 = K+1
//   B  4x16 f32: lane N = lane&15 ; VGPR0 = K=(lane>=16?2:0), VGPR1 = K+1
//   C/D 16x16  : VGPR r -> M = r + (lane>=16?8:0), N = lane&15
template <int K1, int K2, int NLD, bool RELU>
__global__ void sage_gemm_wmma(const float* __restrict__ A1, int lda1,
                               const float* __restrict__ A2, int lda2,
                               const float* __restrict__ B1,
                               const float* __restrict__ B2,
                               const float* __restrict__ bias,
                               float* __restrict__ C, int ldc) {
    constexpr int K = K1 + K2;
    __shared__ float sA[16 * K];
    const int row0 = blockIdx.x * 16;
    const int wave = threadIdx.x >> 5;
    const int lane = threadIdx.x & 31;

    // Cooperative stage of the 16-row A panel (A1 | A2 along K), float4-wide.
    {
        const int r    = threadIdx.x >> 3;   // 16 rows
        const int tcol = threadIdx.x & 7;    // 8 threads per row
        const int grow = row0 + r;
        float4* srow = (float4*)(sA + r * K);
        const float4* a1 = (const float4*)(A1 + (size_t)grow * lda1);
        const float4* a2 = (const float4*)(A2 + (size_t)grow * lda2);
#pragma unroll
        for (int k4 = tcol; k4 < (K1 >> 2); k4 += 8) srow[k4] = a1[k4];
#pragma unroll
        for (int k4 = tcol; k4 < (K2 >> 2); k4 += 8) srow[(K1 >> 2) + k4] = a2[k4];
    }
    __syncthreads();

    const int col  = blockIdx.y * 64 + wave * 16 + (lane & 15);
    const int koff = (lane >> 4) << 1;                 // 0 or 2
    const float* sAb = sA + (lane & 15) * K + koff;    // ds base (imm offsets)
    const float* B1b = B1 + (size_t)koff * NLD + col;  // global bases
    const float* B2b = B2 + (size_t)koff * NLD + col;

    v8f acc = {0.f, 0.f, 0.f, 0.f, 0.f, 0.f, 0.f, 0.f};

#pragma unroll
    for (int k = 0; k < K; k += 4) {
        v2f a, b;
        a.x = sAb[k];
        a.y = sAb[k + 1];
        if (k < K1) {                      // compile-time resolved per chunk
            b.x = B1b[(size_t)k * NLD];
            b.y = B1b[(size_t)(k + 1) * NLD];
        } else {
            b.x = B2b[(size_t)(k - K1) * NLD];
            b.y = B2b[(size_t)(k - K1 + 1) * NLD];
        }
        // (neg_a, A, neg_b, B, c_mod, C, reuse_a, reuse_b)
        acc = __builtin_amdgcn_wmma_f32_16x16x4_f32(
                  false, a, false, b, (short)0, acc, false, false);
    }

    const float bv = bias[col];
    const int rbase = row0 + ((lane >> 4) << 3);
#pragma unroll
    for (int r = 0; r < 8; ++r) {
        float v = acc[r] + bv;
        if (RELU) v = fmaxf(v, 0.0f);
        C[(size_t)(rbase + r) * ldc + col] = v;       // guard-free: padded C
    }
}

// ---------------------------------------------------------------- log_softmax
// One wave32 per row; input has leading dim ldi (=64, padded), N=47 valid.
__global__ void logsoftmax_kernel(const float* __restrict__ in, int ldi,
                                  float* __restrict__ out, int M, int N) {
    int row  = blockIdx.x * (blockDim.x >> 5) + (threadIdx.x >> 5);
    int lane = threadIdx.x & 31;
    if (row >= M) return;
    const float* p = in + (size_t)row * ldi;
    float v0 = (lane      < N) ? p[lane]      : -__builtin_inff();
    float v1 = (lane + 32 < N) ? p[lane + 32] : -__builtin_inff();
    float m = fmaxf(v0, v1);
    for (int off = 16; off > 0; off >>= 1) m = fmaxf(m, __shfl_xor(m, off, 32));
    float s = ((lane      < N) ? expf(v0 - m) : 0.0f) +
              ((lane + 32 < N) ? expf(v1 - m) : 0.0f);
    for (int off = 16; off > 0; off >>= 1) s += __shfl_xor(s, off, 32);
    float ls = logf(s);
    float* q = out + (size_t)row * N;
    if (lane      < N) q[lane]      = v0 - m - ls;
    if (lane + 32 < N) q[lane + 32] = v1 - m - ls;
}

// ------------------------------------------------------------------- launcher
extern "C" void kernel_launch(void* const* d_in, const int* in_sizes, int n_in,
                              void* d_out, int out_size, void* d_ws, size_t ws_size,
                              hipStream_t stream) {
    const float* x    = (const float*)d_in[0];
    const float* W_l0 = (const float*)d_in[1];
    const float* b_l0 = (const float*)d_in[2];
    const float* W_r0 = (const float*)d_in[3];
    const float* W_l1 = (const float*)d_in[4];
    const float* b_l1 = (const float*)d_in[5];
    const float* W_r1 = (const float*)d_in[6];
    const int*   src0 = (const int*)d_in[7];
    const int*   dst0 = (const int*)d_in[8];
    const int*   src1 = (const int*)d_in[9];
    const int*   dst1 = (const int*)d_in[10];
    float*       out  = (float*)d_out;

    // Workspace layout (floats):
    //   agg0 | cnt0 | agg1 | cnt1 | h | pre(pad64) | W_l1p | W_r1p | b_l1p
    float* agg0  = (float*)d_ws;
    float* cnt0  = agg0  + (size_t)N1c * INc;
    float* agg1  = cnt0  + N1c;
    float* cnt1  = agg1  + (size_t)N2c * HIDc;
    float* h     = cnt1  + N2c;
    float* pre   = h     + (size_t)N1c * HIDc;
    float* W_l1p = pre   + (size_t)N2c * OUTP;
    float* W_r1p = W_l1p + (size_t)HIDc * OUTP;
    float* b_l1p = W_r1p + (size_t)HIDc * OUTP;

    // 1) zero the accumulation region (agg0,cnt0,agg1,cnt1 are contiguous)
    size_t zf = (size_t)N1c * INc + N1c + (size_t)N2c * HIDc + N2c; // mult of 4
    int n4 = (int)(zf >> 2);
    zero_f4_kernel<<<(n4 + 255) / 256, 256, 0, stream>>>((float4*)d_ws, n4);

    // 1b) pad layer-1 weights/bias from 47 to 64 columns (guard-free GEMM1)
    pad_cols64_kernel<<<(HIDc * 64 + 255) / 256, 256, 0, stream>>>(W_l1, W_l1p, HIDc, OUTc);
    pad_cols64_kernel<<<(HIDc * 64 + 255) / 256, 256, 0, stream>>>(W_r1, W_r1p, HIDc, OUTc);
    pad_cols64_kernel<<<1, 64, 0, stream>>>(b_l1, b_l1p, 1, OUTc);

    // 2) layer-0 scatter-mean: 32 lanes/edge (IN=128 -> cshift=5, dshift=7)
    scatter_mean_accum<<<(E0c * 32 + 255) / 256, 256, 0, stream>>>(
        x, src0, dst0, agg0, cnt0, E0c, INc, 5);
    div_cnt_kernel<<<(N1c * INc + 255) / 256, 256, 0, stream>>>(
        agg0, cnt0, N1c * INc, 7);

    // 3) layer-0 GEMM: h = relu(agg0@W_l0 + x[:N1]@W_r0 + b_l0)  [50000x256]
    {
        dim3 g(N1c / 16, HIDc / 64);  // 3125 x 4
        sage_gemm_wmma<INc, INc, HIDc, true><<<g, 128, 0, stream>>>(
            agg0, INc, x, INc, W_l0, W_r0, b_l0, h, HIDc);
    }

    // 4) layer-1 scatter-mean: 64 lanes/edge (HID=256 -> cshift=6, dshift=8)
    scatter_mean_accum<<<(E1c * 64 + 255) / 256, 256, 0, stream>>>(
        h, src1, dst1, agg1, cnt1, E1c, HIDc, 6);
    div_cnt_kernel<<<(N2c * HIDc + 255) / 256, 256, 0, stream>>>(
        agg1, cnt1, N2c * HIDc, 8);

    // 5) layer-1 GEMM: pre = agg1@W_l1p + h[:N2]@W_r1p + b_l1p   [4096x64pad]
    {
        dim3 g(N2c / 16, OUTP / 64);  // 256 x 1
        sage_gemm_wmma<HIDc, HIDc, OUTP, false><<<g, 128, 0, stream>>>(
            agg1, HIDc, h, HIDc, W_l1p, W_r1p, b_l1p, pre, OUTP);
    }

    // 6) log_softmax over the 47 valid classes, one wave per row
    logsoftmax_kernel<<<(N2c + 7) / 8, 256, 0, stream>>>(pre, OUTP, out, N2c, OUTc);
}